// NormalizeCLAHE_31258771980570
// MI455X (gfx1250) — compile-verified
//
#include <hip/hip_runtime.h>

typedef __attribute__((ext_vector_type(2))) float v2f;
typedef __attribute__((ext_vector_type(8))) float v8f;

#define NTOT    (16u*1024u*1024u)   // total pixels
#define IMG_H   1024
#define IMG_W   1024
#define GRID_G  8
#define TILE_H  128
#define TILE_W  128
#define NBINS   256
#define NTILES  1024                // 16 images * 8 * 8
#define TPIX    16384               // pixels per tile
#define CLIPV   2560.0f             // int(40.0 * 16384 / 256)

struct Scalars {
    unsigned minEnc;   // ordered-uint encoded min
    unsigned maxEnc;   // ordered-uint encoded max
    double   sum;
    double   sumsq;
    float    A, B;     // fused affine: x3 = A*x + B
};

__device__ __forceinline__ unsigned encF(float f) {
    unsigned u = __float_as_uint(f);
    return (u & 0x80000000u) ? ~u : (u | 0x80000000u);
}
__device__ __forceinline__ float decF(unsigned e) {
    unsigned u = (e & 0x80000000u) ? (e & 0x7FFFFFFFu) : ~e;
    return __uint_as_float(u);
}

__global__ void k_init(Scalars* s) {
    s->minEnc = 0xFFFFFFFFu;   // encoded +inf (atomicMin target)
    s->maxEnc = 0u;            // encoded -inf (atomicMax target)
    s->sum    = 0.0;
    s->sumsq  = 0.0;
}

__global__ void k_reduce(const float* __restrict__ x, Scalars* s) {
    __shared__ float smn[256], smx[256], ssm[256], ssq[256];
    const unsigned tid    = threadIdx.x;
    const unsigned gid    = blockIdx.x * blockDim.x + tid;
    const unsigned stride = gridDim.x * blockDim.x;
    float mn = __builtin_huge_valf(), mx = -__builtin_huge_valf();
    float sm = 0.0f, sq = 0.0f;
    const float4* x4 = (const float4*)x;
    for (unsigned i = gid; i < NTOT / 4u; i += stride) {
        float4 v = x4[i];
        mn = fminf(mn, fminf(fminf(v.x, v.y), fminf(v.z, v.w)));
        mx = fmaxf(mx, fmaxf(fmaxf(v.x, v.y), fmaxf(v.z, v.w)));
        sm += v.x + v.y + v.z + v.w;
        sq += v.x*v.x + v.y*v.y + v.z*v.z + v.w*v.w;
    }
    smn[tid] = mn; smx[tid] = mx; ssm[tid] = sm; ssq[tid] = sq;
    __syncthreads();
    for (int o = 128; o > 0; o >>= 1) {
        if ((int)tid < o) {
            smn[tid] = fminf(smn[tid], smn[tid + o]);
            smx[tid] = fmaxf(smx[tid], smx[tid + o]);
            ssm[tid] += ssm[tid + o];
            ssq[tid] += ssq[tid + o];
        }
        __syncthreads();
    }
    if (tid == 0) {
        atomicMin(&s->minEnc, encF(smn[0]));
        atomicMax(&s->maxEnc, encF(smx[0]));
        atomicAdd(&s->sum,   (double)ssm[0]);
        atomicAdd(&s->sumsq, (double)ssq[0]);
    }
}

__global__ void k_scalars(Scalars* s) {
    const float  mn = decF(s->minEnc), mx = decF(s->maxEnc);
    const double N  = (double)NTOT;
    const float  a1 = 1.0f / (mx - mn);
    const float  b1 = -mn * a1;
    const float  mean  = (float)(s->sum / N);
    const double var   = (s->sumsq - s->sum * s->sum / N) / (N - 1.0);
    const float  mean1 = a1 * mean + b1;
    const float  std1  = (float)sqrt(var) * a1;   // std of x1
    const float  a2 = a1 / std1;
    const float  b2 = (b1 - mean1) / std1;
    const float  mn2 = a2 * mn + b2, mx2 = a2 * mx + b2;
    const float  inv2 = 1.0f / (mx2 - mn2);
    s->A = a2 * inv2;
    s->B = (b2 - mn2) * inv2;
}

__device__ __forceinline__ v8f wmma4(v2f a, v2f b, v8f c) {
    return __builtin_amdgcn_wmma_f32_16x16x4_f32(false, a, false, b, (short)0, c,
                                                 false, false);
}

// One workgroup (256 threads) per tile: LDS histogram -> clip/residual ->
// WMMA-based 256-bin cumsum -> LUT to global scratch.
__global__ void k_hist_lut(const float* __restrict__ x,
                           const Scalars* __restrict__ sc,
                           float* __restrict__ lut) {
    __shared__ unsigned histU[NBINS];
    __shared__ float Hf[NBINS];   // clipped+residual histogram, 16x16 row-major
    __shared__ float Gf[NBINS];   // G = H * J  (broadcast row sums)
    __shared__ float red[NBINS];  // reduction scratch / LUT staging
    const unsigned tid = threadIdx.x;
    const unsigned t   = blockIdx.x;       // tile id
    const unsigned b   = t >> 6;           // image
    const unsigned ti  = (t >> 3) & 7;
    const unsigned tj  = t & 7;
    const float A = sc->A, B = sc->B;

    histU[tid] = 0u;
    __syncthreads();

    const unsigned base = (b * IMG_H + ti * TILE_H) * IMG_W + tj * TILE_W;
    #pragma unroll 4
    for (int i = 0; i < TPIX / 256; ++i) {
        unsigned p  = tid + (unsigned)i * 256u;     // 0..16383 within tile
        unsigned py = p >> 7, px = p & 127u;
        float v = fmaf(A, x[base + py * IMG_W + px], B);
        int bi = (int)(v * 256.0f);
        bi = bi < 0 ? 0 : (bi > 255 ? 255 : bi);
        atomicAdd(&histU[bi], 1u);                  // ds_add_u32
    }
    __syncthreads();

    // clip + residual redistribution
    float h = fminf((float)histU[tid], CLIPV);
    red[tid] = h;
    __syncthreads();
    for (int o = 128; o > 0; o >>= 1) {
        if ((int)tid < o) red[tid] += red[tid + o];
        __syncthreads();
    }
    Hf[tid] = h + ((float)TPIX - red[0]) * (1.0f / (float)NBINS);
    __syncthreads();

    // ---- WMMA cumsum: S = Lstrict*(H*J) + H*U over H[16][16] ----
    // A fragment layout (32b, 16x4): lanes 0-15 M rows w/ K=kh,kh+1; lanes 16-31 K=kh+2,kh+3
    const int lane = (int)(tid & 31u);
    const int mrow = lane & 15;
    const int kh   = (lane >> 4) << 1;    // 0 or 2
    const int ncol = lane & 15;

    v8f G = {};                            // G = H * J (row-sum broadcast)
    #pragma unroll
    for (int kk = 0; kk < 16; kk += 4) {
        v2f a; a.x = Hf[mrow * 16 + kk + kh]; a.y = Hf[mrow * 16 + kk + kh + 1];
        v2f bj; bj.x = 1.0f; bj.y = 1.0f;
        G = wmma4(a, bj, G);
    }
    v8f T = {};                            // T = H * U (within-row inclusive prefix)
    #pragma unroll
    for (int kk = 0; kk < 16; kk += 4) {
        int k0 = kk + kh;
        v2f a; a.x = Hf[mrow * 16 + k0]; a.y = Hf[mrow * 16 + k0 + 1];
        v2f bu; bu.x = (k0 <= ncol) ? 1.0f : 0.0f; bu.y = (k0 + 1 <= ncol) ? 1.0f : 0.0f;
        T = wmma4(a, bu, T);
    }
    if (tid < 32u) {                       // spill G (C/D layout) to LDS for B-layout reload
        #pragma unroll
        for (int i = 0; i < 8; ++i)
            Gf[(i + ((lane >> 4) << 3)) * 16 + ncol] = G[i];
    }
    __syncthreads();
    v8f S = T;                             // S += Lstrict * G
    #pragma unroll
    for (int kk = 0; kk < 16; kk += 4) {
        int k0 = kk + kh;
        v2f a; a.x = (k0 < mrow) ? 1.0f : 0.0f; a.y = (k0 + 1 < mrow) ? 1.0f : 0.0f;
        v2f bg; bg.x = Gf[k0 * 16 + ncol]; bg.y = Gf[(k0 + 1) * 16 + ncol];
        S = wmma4(a, bg, S);
    }
    if (tid < 32u) {                       // LUT = floor(clip(cumsum * 255/16384, 0, 255))
        #pragma unroll
        for (int i = 0; i < 8; ++i) {
            int bin = (i + ((lane >> 4) << 3)) * 16 + ncol;
            red[bin] = floorf(fminf(fmaxf(S[i] * (255.0f / 16384.0f), 0.0f), 255.0f));
        }
    }
    __syncthreads();
    lut[t * NBINS + tid] = red[tid];
}

__global__ void k_out(const float* __restrict__ x,
                      const Scalars* __restrict__ sc,
                      const float* __restrict__ lut,
                      float* __restrict__ out) {
    const unsigned idx = blockIdx.x * 256u + threadIdx.x;
    const unsigned b   = idx >> 20;
    const unsigned rem = idx & 0xFFFFFu;
    const unsigned i   = rem >> 10, j = rem & 1023u;
    float v = fmaf(sc->A, x[idx], sc->B);
    int bi = (int)(v * 256.0f);
    bi = bi < 0 ? 0 : (bi > 255 ? 255 : bi);
    float gi = fminf(fmaxf(((float)i + 0.5f) * (1.0f / 128.0f) - 0.5f, 0.0f), 7.0f);
    float gj = fminf(fmaxf(((float)j + 0.5f) * (1.0f / 128.0f) - 0.5f, 0.0f), 7.0f);
    int i0 = (int)gi; int i1 = i0 + 1 > 7 ? 7 : i0 + 1; float wy = gi - (float)i0;
    int j0 = (int)gj; int j1 = j0 + 1 > 7 ? 7 : j0 + 1; float wx = gj - (float)j0;
    const float* lb = lut + (size_t)b * 64u * NBINS;
    float v00 = lb[(i0 * 8 + j0) * NBINS + bi];
    float v01 = lb[(i0 * 8 + j1) * NBINS + bi];
    float v10 = lb[(i1 * 8 + j0) * NBINS + bi];
    float v11 = lb[(i1 * 8 + j1) * NBINS + bi];
    float o = (1.0f - wy) * ((1.0f - wx) * v00 + wx * v01) +
              wy         * ((1.0f - wx) * v10 + wx * v11);
    out[idx] = o * (1.0f / 255.0f);
}

__global__ void k_ycopy(const unsigned char* __restrict__ y,
                        unsigned char* __restrict__ dst, int nbytes) {
    int i = (int)threadIdx.x;
    if (i < nbytes) dst[i] = y[i];
}

extern "C" void kernel_launch(void* const* d_in, const int* in_sizes, int n_in,
                              void* d_out, int out_size, void* d_ws, size_t ws_size,
                              hipStream_t stream) {
    const float* x  = (const float*)d_in[0];
    Scalars*     sc = (Scalars*)d_ws;
    float*       lut = (float*)((char*)d_ws + 64);          // 1024*256 floats = 1 MB
    float*       out = (float*)d_out;

    k_init    <<<1, 1, 0, stream>>>(sc);
    k_reduce  <<<1024, 256, 0, stream>>>(x, sc);
    k_scalars <<<1, 1, 0, stream>>>(sc);
    k_hist_lut<<<NTILES, 256, 0, stream>>>(x, sc, lut);
    k_out     <<<NTOT / 256u, 256, 0, stream>>>(x, sc, lut, out);

    // y passthrough: raw-byte copy into the output tail (if space was allotted)
    long long tail_bytes = ((long long)out_size - (long long)NTOT) * 4ll;
    if (n_in > 1 && tail_bytes > 0) {
        int nb = tail_bytes < 128 ? (int)tail_bytes : 128;
        k_ycopy<<<1, 256, 0, stream>>>((const unsigned char*)d_in[1],
                                       (unsigned char*)d_out + (size_t)NTOT * 4u, nb);
    }
}